// DecoderLayer_70729521430581
// MI455X (gfx1250) — compile-verified
//
#include <hip/hip_runtime.h>

typedef __attribute__((ext_vector_type(2))) float v2f;
typedef __attribute__((ext_vector_type(8))) float v8f;

#define D_MODEL 2048
#define NHEAD   16
#define DHEAD   128
#define QLEN    16
#define PLEN    8192
#define KLEN    8208
#define FDIM    5632
#define NROWS   32   // B*Q
#define NBH     32   // B*H
#define N_PRUNE 820  // int(0.1 * 8208)
#define NSEG    4    // key-dim segments for attn@V (each 2052 keys, 4-aligned)
#define SEGLEN  2052

// ---------------- RMSNorm: one block per row ----------------
__global__ void rmsnorm_kernel(const float* __restrict__ x, const float* __restrict__ w,
                               float* __restrict__ out) {
  __shared__ float red[256];
  const int row = blockIdx.x;
  const float* xr = x + (size_t)row * D_MODEL;
  float s = 0.f;
  for (int i = threadIdx.x; i < D_MODEL; i += 256) { float v = xr[i]; s += v * v; }
  red[threadIdx.x] = s; __syncthreads();
  for (int o = 128; o > 0; o >>= 1) {
    if (threadIdx.x < o) red[threadIdx.x] += red[threadIdx.x + o];
    __syncthreads();
  }
  const float scale = rsqrtf(red[0] / (float)D_MODEL + 1e-5f);
  for (int i = threadIdx.x; i < D_MODEL; i += 256)
    out[(size_t)row * D_MODEL + i] = xr[i] * scale * w[i];
}

// ---------------- 32xN fp32 WMMA GEMM: C = A@B (+R) ----------------
// One wave per 16-wide N tile; BOTH 16-row M tiles computed per wave so each
// B element is loaded exactly once (weights stream once from HBM).
__global__ __launch_bounds__(32) void wmma_gemm_kernel(
    const float* __restrict__ A, int lda,
    const float* __restrict__ B, int ldb,
    float* __restrict__ C, int ldc,
    const float* __restrict__ R, int Kdim)
{
  const int lane = threadIdx.x & 31;
  const int half = lane >> 4, l16 = lane & 15;
  const int nbase = blockIdx.x * 16;
  const float* a0 = A + (size_t)l16 * lda + half * 2;          // rows 0..15
  const float* a1 = A + (size_t)(16 + l16) * lda + half * 2;   // rows 16..31
  const float* b0 = B + (size_t)(half * 2) * ldb + nbase + l16;
  v8f acc0 = {}, acc1 = {};
  for (int k = 0; k < Kdim; k += 4) {
    __builtin_prefetch(b0 + (size_t)(k + 32) * ldb, 0, 1);     // global_prefetch_b8
    v2f bv, av0, av1;
    bv.x  = b0[(size_t)k * ldb];  bv.y  = b0[(size_t)(k + 1) * ldb];
    av0.x = a0[k];                av0.y = a0[k + 1];
    av1.x = a1[k];                av1.y = a1[k + 1];
    acc0 = __builtin_amdgcn_wmma_f32_16x16x4_f32(false, av0, false, bv, (short)0, acc0, false, false);
    acc1 = __builtin_amdgcn_wmma_f32_16x16x4_f32(false, av1, false, bv, (short)0, acc1, false, false);
  }
#pragma unroll
  for (int r = 0; r < 8; ++r) {
    const int m = half * 8 + r, n = nbase + l16;
    float v0 = acc0[r], v1 = acc1[r];
    if (R) { v0 += R[(size_t)m * ldc + n]; v1 += R[(size_t)(m + 16) * ldc + n]; }
    C[(size_t)m * ldc + n]        = v0;
    C[(size_t)(m + 16) * ldc + n] = v1;
  }
}

// ---------------- RoPE + [row,d] -> [b,h,q,dh] reshape for q,k,v ----------------
__global__ void rope_reshape_kernel(const float* __restrict__ qp, const float* __restrict__ kp,
                                    const float* __restrict__ vp,
                                    float* __restrict__ qr, float* __restrict__ kr,
                                    float* __restrict__ vt)
{
  const int idx = blockIdx.x * blockDim.x + threadIdx.x;
  if (idx >= NROWS * D_MODEL) return;
  const int row = idx / D_MODEL, col = idx % D_MODEL;
  const int b = row / QLEN, qi = row % QLEN;
  const int h = col / DHEAD, dh = col % DHEAD;
  const int fi = dh & 63;
  const float inv_freq = __expf(-(2.0f * (float)fi / (float)DHEAD) * __logf(10000.0f));
  const float ang = (float)(PLEN + qi) * inv_freq;
  const float c = __cosf(ang), s = __sinf(ang);
  const size_t src = (size_t)row * D_MODEL + col;
  const size_t prt = (size_t)row * D_MODEL + (dh < 64 ? col + 64 : col - 64);
  const float sgn = (dh < 64) ? -1.0f : 1.0f;
  const size_t dst = (((size_t)(b * NHEAD + h)) * QLEN + qi) * DHEAD + dh;
  qr[dst] = qp[src] * c + sgn * qp[prt] * s;
  kr[dst] = kp[src] * c + sgn * kp[prt] * s;
  vt[dst] = vp[src];
}

// ---------------- scores[bh,q,k] = (q . k) / sqrt(Dh) via WMMA ----------------
__global__ __launch_bounds__(32) void attn_scores_kernel(
    const float* __restrict__ qr, const float* __restrict__ pk, const float* __restrict__ kr,
    float* __restrict__ sc)
{
  const int lane = threadIdx.x & 31, half = lane >> 4, l16 = lane & 15;
  const int bh = blockIdx.y;
  const int nbase = blockIdx.x * 16;          // key tile (P=8192 is tile-aligned)
  const float* aq = qr + ((size_t)bh * QLEN + l16) * DHEAD + half * 2;
  const float* kb = (nbase < PLEN)
      ? pk + ((size_t)bh * PLEN + nbase + l16) * DHEAD + half * 2
      : kr + ((size_t)bh * QLEN + (nbase - PLEN) + l16) * DHEAD + half * 2;
  v8f acc = {};
  for (int k = 0; k < DHEAD; k += 4) {
    v2f av, bv;
    av.x = aq[k]; av.y = aq[k + 1];
    bv.x = kb[k]; bv.y = kb[k + 1];
    acc = __builtin_amdgcn_wmma_f32_16x16x4_f32(false, av, false, bv, (short)0, acc, false, false);
  }
  const float scale = 0.08838834764831845f;   // 1/sqrt(128)
#pragma unroll
  for (int r = 0; r < 8; ++r) {
    const int m = half * 8 + r, n = nbase + l16;
    sc[((size_t)bh * QLEN + m) * KLEN + n] = acc[r] * scale;
  }
}

// ---------------- causal-masked softmax, in place over d_out attn slice ----------------
__global__ void softmax_kernel(float* __restrict__ attn)
{
  __shared__ float red[256];
  const int row = blockIdx.x;                 // bh*16 + q
  const int qi = row & 15;
  const int limit = PLEN + qi;                // inclusive visible key pos
  float* a = attn + (size_t)row * KLEN;
  float mx = -3.4e38f;
  for (int k = threadIdx.x; k < KLEN; k += 256) {
    const float v = a[k] + ((k <= limit) ? 0.f : -1e9f);
    mx = fmaxf(mx, v);
  }
  red[threadIdx.x] = mx; __syncthreads();
  for (int o = 128; o > 0; o >>= 1) {
    if (threadIdx.x < o) red[threadIdx.x] = fmaxf(red[threadIdx.x], red[threadIdx.x + o]);
    __syncthreads();
  }
  mx = red[0]; __syncthreads();
  float sm = 0.f;
  for (int k = threadIdx.x; k < KLEN; k += 256) {
    const float v = a[k] + ((k <= limit) ? 0.f : -1e9f);
    sm += __expf(v - mx);
  }
  red[threadIdx.x] = sm; __syncthreads();
  for (int o = 128; o > 0; o >>= 1) {
    if (threadIdx.x < o) red[threadIdx.x] += red[threadIdx.x + o];
    __syncthreads();
  }
  const float inv = 1.f / red[0];
  for (int k = threadIdx.x; k < KLEN; k += 256) {
    const float v = a[k] + ((k <= limit) ? 0.f : -1e9f);
    a[k] = __expf(v - mx) * inv;
  }
}

// ---------------- partial attn @ v_full via WMMA ----------------
// One wave per (bh, key-segment); the wave holds all 8 dh-tiles (Dh=128) so
// attn rows and V rows are each streamed exactly once per segment.
__global__ __launch_bounds__(32) void attn_v_kernel(
    const float* __restrict__ attn, const float* __restrict__ pv, const float* __restrict__ vt,
    float* __restrict__ part)
{
  const int lane = threadIdx.x & 31, half = lane >> 4, l16 = lane & 15;
  const int bh = blockIdx.y;
  const int seg = blockIdx.x;                 // 4 segments of 2052 keys
  const int b = bh / NHEAD, h = bh % NHEAD;
  const int k0 = seg * SEGLEN, k1 = k0 + SEGLEN;
  const float* ar = attn + ((size_t)bh * QLEN + l16) * KLEN + half * 2;
  v8f acc[8];
#pragma unroll
  for (int t = 0; t < 8; ++t) acc[t] = (v8f){};
  for (int k = k0; k < k1; k += 4) {
    const int kk = k + half * 2;
    const float* vr = (k < PLEN) ? pv + ((size_t)bh * PLEN + kk) * DHEAD
                                 : vt + ((size_t)bh * QLEN + (kk - PLEN)) * DHEAD;
    v2f av; av.x = ar[k]; av.y = ar[k + 1];
#pragma unroll
    for (int t = 0; t < 8; ++t) {
      v2f bv; bv.x = vr[t * 16 + l16]; bv.y = vr[DHEAD + t * 16 + l16];
      acc[t] = __builtin_amdgcn_wmma_f32_16x16x4_f32(false, av, false, bv, (short)0, acc[t], false, false);
    }
  }
#pragma unroll
  for (int t = 0; t < 8; ++t)
#pragma unroll
    for (int r = 0; r < 8; ++r) {
      const int m = half * 8 + r;
      part[((size_t)seg * NROWS + b * QLEN + m) * D_MODEL + h * DHEAD + t * 16 + l16] = acc[t][r];
    }
}

// ---------------- sum the NSEG partial context buffers ----------------
__global__ void seg_reduce_kernel(const float* __restrict__ part, float* __restrict__ ctx)
{
  const int i = blockIdx.x * blockDim.x + threadIdx.x;
  if (i >= NROWS * D_MODEL) return;
  float s = 0.f;
#pragma unroll
  for (int seg = 0; seg < NSEG; ++seg)
    s += part[(size_t)seg * NROWS * D_MODEL + i];
  ctx[i] = s;
}

// ---------------- gu = silu(gate) * up, in place into gate buffer ----------------
__global__ void silu_mul_kernel(float* __restrict__ g, const float* __restrict__ u, int n)
{
  const int i = blockIdx.x * blockDim.x + threadIdx.x;
  if (i < n) { const float x = g[i]; g[i] = (x / (1.f + __expf(-x))) * u[i]; }
}

// ---------------- imp[k] = mean over (b,h) of attn[:, :, Q-1, k]; imp[K-1]=inf ----------------
__global__ void importance_kernel(const float* __restrict__ attn, float* __restrict__ imp)
{
  const int k = blockIdx.x * blockDim.x + threadIdx.x;
  if (k >= KLEN) return;
  float s = 0.f;
  for (int bh = 0; bh < NBH; ++bh)
    s += attn[((size_t)bh * QLEN + (QLEN - 1)) * KLEN + k];
  imp[k] = (k == KLEN - 1) ? __builtin_inff() : s / (float)NBH;
}

// ---------------- top_k(-imp, 820): 820 smallest, lowest index on ties ----------------
__global__ void select_prune_kernel(const float* __restrict__ imp, int* __restrict__ out_idx)
{
  __shared__ float sval[256];
  __shared__ int   sidx[256];
  __shared__ unsigned char taken[KLEN];
  for (int k = threadIdx.x; k < KLEN; k += 256) taken[k] = 0;
  __syncthreads();
  for (int it = 0; it < N_PRUNE; ++it) {
    float bv = 3.4e38f; int bi = KLEN;
    for (int k = threadIdx.x; k < KLEN; k += 256) {
      if (!taken[k]) {
        const float v = imp[k];
        if (v < bv || (v == bv && k < bi)) { bv = v; bi = k; }
      }
    }
    sval[threadIdx.x] = bv; sidx[threadIdx.x] = bi;
    __syncthreads();
    for (int o = 128; o > 0; o >>= 1) {
      if (threadIdx.x < o) {
        const float v2 = sval[threadIdx.x + o]; const int i2 = sidx[threadIdx.x + o];
        if (v2 < sval[threadIdx.x] || (v2 == sval[threadIdx.x] && i2 < sidx[threadIdx.x])) {
          sval[threadIdx.x] = v2; sidx[threadIdx.x] = i2;
        }
      }
      __syncthreads();
    }
    if (threadIdx.x == 0) { out_idx[it] = sidx[0]; taken[sidx[0]] = 1; }
    __syncthreads();
  }
}

extern "C" void kernel_launch(void* const* d_in, const int* in_sizes, int n_in,
                              void* d_out, int out_size, void* d_ws, size_t ws_size,
                              hipStream_t stream)
{
  (void)in_sizes; (void)n_in; (void)out_size; (void)ws_size;
  const float* hidden  = (const float*)d_in[0];
  const float* past_k  = (const float*)d_in[1];
  const float* past_v  = (const float*)d_in[2];
  const float* wq      = (const float*)d_in[3];
  const float* wk      = (const float*)d_in[4];
  const float* wv      = (const float*)d_in[5];
  const float* wo      = (const float*)d_in[6];
  const float* w_gate  = (const float*)d_in[7];
  const float* w_up    = (const float*)d_in[8];
  const float* w_down  = (const float*)d_in[9];
  const float* norm1_w = (const float*)d_in[10];
  const float* norm2_w = (const float*)d_in[11];
  // d_in[12] = attn_mask (recomputed analytically on device)

  float* out        = (float*)d_out;
  float* new_hidden = out;                                   // 32*2048  = 65536
  float* attn       = out + 65536;                           // 32*16*K  = 4202496
  int*   prune_idx  = (int*)d_out + 65536 + 4202496;         // 820 ints

  float* ws   = (float*)d_ws;
  float* xn   = ws;                 // 65536
  float* qp   = ws + 65536;         // 65536
  float* kp   = ws + 131072;        // 65536
  float* vp   = ws + 196608;        // 65536
  float* qr   = ws + 262144;        // 65536
  float* kr   = ws + 327680;        // 65536
  float* vt   = ws + 393216;        // 65536
  float* ctx  = ws + 458752;        // 65536
  float* hb   = ws + 524288;        // 65536
  float* yn   = ws + 589824;        // 65536
  float* g    = ws + 655360;        // 180224
  float* u    = ws + 835584;        // 180224
  float* imp  = ws + 1015808;       // 8208
  float* part = ws + 1024016;       // NSEG*65536 = 262144

  rmsnorm_kernel<<<NROWS, 256, 0, stream>>>(hidden, norm1_w, xn);

  wmma_gemm_kernel<<<D_MODEL / 16, 32, 0, stream>>>(xn, D_MODEL, wq, D_MODEL, qp, D_MODEL, nullptr, D_MODEL);
  wmma_gemm_kernel<<<D_MODEL / 16, 32, 0, stream>>>(xn, D_MODEL, wk, D_MODEL, kp, D_MODEL, nullptr, D_MODEL);
  wmma_gemm_kernel<<<D_MODEL / 16, 32, 0, stream>>>(xn, D_MODEL, wv, D_MODEL, vp, D_MODEL, nullptr, D_MODEL);

  rope_reshape_kernel<<<(NROWS * D_MODEL + 255) / 256, 256, 0, stream>>>(qp, kp, vp, qr, kr, vt);

  attn_scores_kernel<<<dim3(KLEN / 16, NBH), 32, 0, stream>>>(qr, past_k, kr, attn);
  softmax_kernel<<<NBH * QLEN, 256, 0, stream>>>(attn);
  attn_v_kernel<<<dim3(NSEG, NBH), 32, 0, stream>>>(attn, past_v, vt, part);
  seg_reduce_kernel<<<(NROWS * D_MODEL + 255) / 256, 256, 0, stream>>>(part, ctx);

  wmma_gemm_kernel<<<D_MODEL / 16, 32, 0, stream>>>(ctx, D_MODEL, wo, D_MODEL, hb, D_MODEL, hidden, D_MODEL);

  rmsnorm_kernel<<<NROWS, 256, 0, stream>>>(hb, norm2_w, yn);

  wmma_gemm_kernel<<<FDIM / 16, 32, 0, stream>>>(yn, D_MODEL, w_gate, FDIM, g, FDIM, nullptr, D_MODEL);
  wmma_gemm_kernel<<<FDIM / 16, 32, 0, stream>>>(yn, D_MODEL, w_up,   FDIM, u, FDIM, nullptr, D_MODEL);
  silu_mul_kernel<<<(NROWS * FDIM + 255) / 256, 256, 0, stream>>>(g, u, NROWS * FDIM);
  wmma_gemm_kernel<<<D_MODEL / 16, 32, 0, stream>>>(g, FDIM, w_down, D_MODEL, new_hidden, D_MODEL, hb, FDIM);

  importance_kernel<<<(KLEN + 255) / 256, 256, 0, stream>>>(attn, imp);
  select_prune_kernel<<<1, 256, 0, stream>>>(imp, prune_idx);
}